// VectorQuantizeEMA_78658031059236
// MI455X (gfx1250) — compile-verified
//
#include <hip/hip_runtime.h>
#include <hip/hip_bf16.h>

// ---------------------------------------------------------------------------
// VQ-VAE EMA vector quantizer for MI455X (gfx1250, wave32, WMMA)
//   z_e:          [16,4096,64] f32   (T=65536 tokens, E=64)
//   embed:        [1024,64]    f32   (K=1024 codes)
//   cluster_size: [1024]       f32
//   embed_avg:    [1024,64]    f32
// Outputs (flat concat): z_q_st[T*64], diff[1], embed_ind[T] (as float),
//                        new_embed[K*64], new_cluster_size[K], new_embed_avg[K*64]
// ---------------------------------------------------------------------------

typedef __attribute__((ext_vector_type(2))) float v2f;
typedef __attribute__((ext_vector_type(8))) float v8f;

#define TOKENS     65536
#define K_CODES    1024
#define E_DIM      64
#define LDS_STRIDE 68            // 64 + 4 pad -> conflict-free ds_load_b64
#define DECAY      0.99f
#define ONE_M_D    0.01f
#define EPS_F      1e-5f

// Workspace layout (floats)
#define WS_ONEHOT  0                       // 1024
#define WS_ESUM    1024                    // 65536
#define WS_DIFF    (1024 + 65536)          // 1
#define WS_CS      (1024 + 65536 + 1)      // 1024
#define WS_ZERO_FLOATS (1024 + 65536 + 1)  // region that must start at 0

// Output offsets (floats)
#define OUT_ZQ    0
#define OUT_DIFF  4194304
#define OUT_IND   4194305
#define OUT_NE    (4194305 + 65536)
#define OUT_NCS   (OUT_NE + 65536)
#define OUT_NEA   (OUT_NCS + 1024)

// LDS: embed (1024*68) + bias (1024) + token indices (128 ints)
#define LDS_EMBED_F  (K_CODES * LDS_STRIDE)
#define LDS_BIAS_F   K_CODES
#define LDS_IDX_F    128
#define LDS_TOTAL_B  ((LDS_EMBED_F + LDS_BIAS_F + LDS_IDX_F) * 4)

__global__ __launch_bounds__(256)
void vq_main_kernel(const float* __restrict__ z_e,
                    const float* __restrict__ embed,
                    float* __restrict__ out_zq,
                    float* __restrict__ out_ind,
                    float* __restrict__ ws_onehot,
                    float* __restrict__ ws_esum,
                    float* __restrict__ ws_diff)
{
    extern __shared__ float lds[];
    float* lEmbed = lds;                       // [1024][68]
    float* lBias  = lds + LDS_EMBED_F;         // [1024]  = -0.5*||w_n||^2
    int*   lIdx   = (int*)(lds + LDS_EMBED_F + LDS_BIAS_F); // [128]

    const int tid = threadIdx.x;

    // ---- Stage codebook into LDS (padded rows) and compute -0.5*||w||^2 ----
    #pragma unroll
    for (int rr = 0; rr < 4; ++rr) {
        const int row = tid * 4 + rr;                       // 0..1023
        const float4* src = (const float4*)(embed + row * E_DIM);
        float ss = 0.0f;
        #pragma unroll
        for (int j = 0; j < 16; ++j) {
            float4 v = src[j];
            *(float4*)&lEmbed[row * LDS_STRIDE + 4 * j] = v;
            ss += v.x * v.x + v.y * v.y + v.z * v.z + v.w * v.w;
        }
        lBias[row] = -0.5f * ss;
    }
    __syncthreads();

    const int lane = tid & 31;
    const int wave = tid >> 5;
    const int half = lane >> 4;       // 0: K even pair / rows 0-7 ; 1: +2 / rows 8-15
    const int lm   = lane & 15;
    const int tokBase = (blockIdx.x * 8 + wave) * 16;

    // ---- Preload A block: 16 tokens x 64 dims -> 16 K-step operands ----
    // A layout (16x4 f32): lane L holds row m=L%16; VGPR0 = K (4k+2h), VGPR1 = K+1
    v2f a[16];
    const float* arow = z_e + (size_t)(tokBase + lm) * E_DIM + 2 * half;
    #pragma unroll
    for (int k = 0; k < 16; ++k)
        a[k] = *(const v2f*)(arow + 4 * k);

    // ---- Sweep 64 code tiles: score = x.w - 0.5||w||^2, track argmax ----
    float bestv[8];
    int   besti[8];
    #pragma unroll
    for (int r = 0; r < 8; ++r) { bestv[r] = -3.0e38f; besti[r] = 0; }

    for (int tile = 0; tile < 64; ++tile) {
        const int n = tile * 16 + lm;              // C/D layout: n = lane%16
        const float bias = lBias[n];
        v8f c = { bias, bias, bias, bias, bias, bias, bias, bias };
        const float* brow = lEmbed + n * LDS_STRIDE + 2 * half;
        #pragma unroll
        for (int k = 0; k < 16; ++k) {
            v2f b = *(const v2f*)(brow + 4 * k);   // ds_load_b64, bank-conflict-free
            c = __builtin_amdgcn_wmma_f32_16x16x4_f32(
                    false, a[k], false, b, (short)0, c, false, false);
        }
        #pragma unroll
        for (int r = 0; r < 8; ++r) {
            if (c[r] > bestv[r]) { bestv[r] = c[r]; besti[r] = n; }
        }
    }

    // ---- Cross-lane argmax within 16-lane groups (rows m = r + 8*half) ----
    #pragma unroll
    for (int r = 0; r < 8; ++r) {
        float v = bestv[r];
        int   i = besti[r];
        #pragma unroll
        for (int off = 1; off < 16; off <<= 1) {
            float ov = __shfl_xor(v, off, 32);
            int   oi = __shfl_xor(i, off, 32);
            if (ov > v || (ov == v && oi < i)) { v = ov; i = oi; }
        }
        besti[r] = i;
    }
    if (lm == 0) {
        #pragma unroll
        for (int r = 0; r < 8; ++r)
            lIdx[wave * 16 + half * 8 + r] = besti[r];
    }
    __syncthreads();

    // ---- Emit z_q (gather from LDS codebook), diff, indices, EMA scatters ----
    float dsum = 0.0f;
    for (int t = 0; t < 16; ++t) {
        const int tok = tokBase + t;
        const int idx = lIdx[wave * 16 + t];
        float2 q = *(const float2*)&lEmbed[idx * LDS_STRIDE + 2 * lane];
        float2 z = *(const float2*)(z_e + (size_t)tok * E_DIM + 2 * lane);
        *(float2*)(out_zq + (size_t)tok * E_DIM + 2 * lane) = q;
        float dx = q.x - z.x, dy = q.y - z.y;
        dsum += dx * dx + dy * dy;
        atomicAdd(&ws_esum[idx * E_DIM + 2 * lane],     z.x);
        atomicAdd(&ws_esum[idx * E_DIM + 2 * lane + 1], z.y);
        if (lane == 0) {
            atomicAdd(&ws_onehot[idx], 1.0f);
            out_ind[tok] = (float)idx;
        }
    }
    #pragma unroll
    for (int off = 1; off < 32; off <<= 1)
        dsum += __shfl_xor(dsum, off, 32);
    if (lane == 0)
        atomicAdd(ws_diff, dsum);
}

// ---- Finalize A: new_cluster_size, normalization cs, scalar diff ----
__global__ __launch_bounds__(1024)
void vq_final_a_kernel(const float* __restrict__ cluster_size,
                       const float* __restrict__ ws_onehot,
                       const float* __restrict__ ws_diff,
                       float* __restrict__ out_ncs,
                       float* __restrict__ out_diff,
                       float* __restrict__ ws_cs)
{
    __shared__ float red[1024];
    const int k = threadIdx.x;
    const float ncs = DECAY * cluster_size[k] + ONE_M_D * ws_onehot[k];
    out_ncs[k] = ncs;
    red[k] = ncs;
    __syncthreads();
    for (int s = 512; s > 0; s >>= 1) {
        if (k < s) red[k] += red[k + s];
        __syncthreads();
    }
    const float n = red[0];
    ws_cs[k] = (ncs + EPS_F) / (n + (float)K_CODES * EPS_F) * n;
    if (k == 0)
        *out_diff = *ws_diff / ((float)TOKENS * (float)E_DIM);
}

// ---- Finalize B: new_embed_avg and new_embed ----
__global__ __launch_bounds__(256)
void vq_final_b_kernel(const float* __restrict__ embed_avg,
                       const float* __restrict__ ws_esum,
                       const float* __restrict__ ws_cs,
                       float* __restrict__ out_nea,
                       float* __restrict__ out_ne)
{
    const int i = blockIdx.x * 256 + threadIdx.x;   // < 65536
    const float nea = DECAY * embed_avg[i] + ONE_M_D * ws_esum[i];
    out_nea[i] = nea;
    out_ne[i]  = nea / ws_cs[i >> 6];
}

extern "C" void kernel_launch(void* const* d_in, const int* in_sizes, int n_in,
                              void* d_out, int out_size, void* d_ws, size_t ws_size,
                              hipStream_t stream)
{
    const float* z_e          = (const float*)d_in[0];
    const float* embed        = (const float*)d_in[1];
    const float* cluster_size = (const float*)d_in[2];
    const float* embed_avg    = (const float*)d_in[3];

    float* out = (float*)d_out;
    float* ws  = (float*)d_ws;

    // Zero the accumulation region (onehot + esum + diff) every call.
    hipMemsetAsync(d_ws, 0, (size_t)WS_ZERO_FLOATS * sizeof(float), stream);

    static const size_t ldsBytes = LDS_TOTAL_B;     // ~283 KB (<320 KB WGP LDS)
    hipFuncSetAttribute((const void*)vq_main_kernel,
                        hipFuncAttributeMaxDynamicSharedMemorySize, (int)ldsBytes);

    vq_main_kernel<<<TOKENS / 128, 256, ldsBytes, stream>>>(
        z_e, embed,
        out + OUT_ZQ, out + OUT_IND,
        ws + WS_ONEHOT, ws + WS_ESUM, ws + WS_DIFF);

    vq_final_a_kernel<<<1, 1024, 0, stream>>>(
        cluster_size, ws + WS_ONEHOT, ws + WS_DIFF,
        out + OUT_NCS, out + OUT_DIFF, ws + WS_CS);

    vq_final_b_kernel<<<(K_CODES * E_DIM) / 256, 256, 0, stream>>>(
        embed_avg, ws + WS_ESUM, ws + WS_CS,
        out + OUT_NEA, out + OUT_NE);
}